// CrossAttention_24584392802392
// MI455X (gfx1250) — compile-verified
//
#include <hip/hip_runtime.h>
#include <math.h>
#include <stdint.h>

typedef float v2f __attribute__((ext_vector_type(2)));
typedef float v4f __attribute__((ext_vector_type(4)));
typedef float v8f __attribute__((ext_vector_type(8)));

#define BB 4
#define HH 2
#define LL 2048
#define DD 16
#define DH 8
#define QT 16
#define NWAVE 8
#define NTHR 256
#define LPAD (LL + 4)   // row stride 2052 floats = 8208 B: 16B-aligned, de-aliases banks

// ---------------------------------------------------------------------------
// Kernel 1: LayerNorm + Linear projection for q/k/v. One thread per row.
// Q rows are pre-scaled by 1/sqrt(dh).
// ---------------------------------------------------------------------------
__global__ __launch_bounds__(NTHR)
void prep_kernel(const float* __restrict__ query, const float* __restrict__ key,
                 const float* __restrict__ value,
                 const float* __restrict__ ln_qg, const float* __restrict__ ln_qb,
                 const float* __restrict__ ln_kg, const float* __restrict__ ln_kb,
                 const float* __restrict__ ln_vg, const float* __restrict__ ln_vb,
                 const float* __restrict__ Wq, const float* __restrict__ bq,
                 const float* __restrict__ Wk, const float* __restrict__ bk,
                 const float* __restrict__ Wv, const float* __restrict__ bv,
                 float* __restrict__ Qs, float* __restrict__ Ks,
                 float* __restrict__ Vs)
{
    __shared__ float sW[256];
    __shared__ float sb[16], sg[16], sbeta[16];

    const int tid   = threadIdx.x;
    const int which = blockIdx.x >> 5;               // 32 blocks per tensor
    const float* X    = which == 0 ? query : (which == 1 ? key : value);
    const float* W    = which == 0 ? Wq    : (which == 1 ? Wk  : Wv);
    const float* bias = which == 0 ? bq    : (which == 1 ? bk  : bv);
    const float* g    = which == 0 ? ln_qg : (which == 1 ? ln_kg : ln_vg);
    const float* beta = which == 0 ? ln_qb : (which == 1 ? ln_kb : ln_vb);
    float* O          = which == 0 ? Qs    : (which == 1 ? Ks  : Vs);

    sW[tid] = W[tid];
    if (tid < 16) { sb[tid] = bias[tid]; sg[tid] = g[tid]; sbeta[tid] = beta[tid]; }
    __syncthreads();

    const int row = (blockIdx.x & 31) * NTHR + tid;  // 0..8191 = b*L + t
    const float* xr = X + (size_t)row * DD;
    float x[16];
    float m = 0.f;
#pragma unroll
    for (int i = 0; i < 16; ++i) { x[i] = xr[i]; m += x[i]; }
    m *= (1.f / 16.f);
    float v = 0.f;
#pragma unroll
    for (int i = 0; i < 16; ++i) { float d = x[i] - m; v += d * d; }
    v *= (1.f / 16.f);
    const float rinv = rsqrtf(v + 1e-5f);
#pragma unroll
    for (int i = 0; i < 16; ++i) x[i] = (x[i] - m) * rinv * sg[i] + sbeta[i];

    const float scale = (which == 0) ? 0.3535533905932738f : 1.f;  // 1/sqrt(8)
    float* orow = O + (size_t)row * DD;
#pragma unroll
    for (int j = 0; j < 16; ++j) {
        float acc = sb[j];
#pragma unroll
        for (int i = 0; i < 16; ++i) acc += x[i] * sW[j * 16 + i];
        orow[j] = acc * scale;
    }
}

// ---------------------------------------------------------------------------
// Kernel 2: attention. One workgroup per (b, h, 16-query tile).
// Full 16 x 2048 f32 score block in LDS: single pass over w/mask, attn
// written exactly once (async, straight from LDS).
// ---------------------------------------------------------------------------
__global__ __launch_bounds__(NTHR)
void attn_kernel(const float* __restrict__ Qs, const float* __restrict__ Ks,
                 const float* __restrict__ Vs,
                 const float* __restrict__ wsrc, const float* __restrict__ msk,
                 float* __restrict__ attn_out, float* __restrict__ ctxws)
{
    __shared__ float s_scores[QT][LPAD];        // ~128 KB
    __shared__ float s_q[QT][DH];
    __shared__ float s_red[QT][16];
    __shared__ float s_rowstat[QT];
    __shared__ float s_ctx[NWAVE][QT][DH];

    const int tid  = threadIdx.x;
    const int lane = tid & 31;
    const int wave = tid >> 5;
    const int qt   = blockIdx.x;     // 0..127
    const int h    = blockIdx.y;
    const int b    = blockIdx.z;
    const int q0   = qt * QT;

    // Load Q tile (already LayerNorm'd, projected, scaled)
    if (tid < QT * DH) {
        const int mrow = tid >> 3, d = tid & 7;
        s_q[mrow][d] = Qs[((size_t)b * LL + q0 + mrow) * DD + h * DH + d];
    }
    __syncthreads();

    // Per-lane WMMA operand positions (ISA 16x4 f32 A layout):
    //   lanes 0-15: K = {0,1}; lanes 16-31: K = {2,3}
    const int Mn  = lane & 15;
    const int kb2 = (lane >> 4) * 2;
    const int hi  = lane >> 4;

    v2f a0, a1;
    a0.x = s_q[Mn][kb2];     a0.y = s_q[Mn][kb2 + 1];
    a1.x = s_q[Mn][4 + kb2]; a1.y = s_q[Mn][4 + kb2 + 1];

    const size_t wbase = (((size_t)b * HH + h) * LL + q0) * LL;

    // -------- Phase 1: raw scores = Q K^T via WMMA, store into LDS
    for (int kt = wave; kt < LL / 16; kt += NWAVE) {
        const int gN = kt * 16 + Mn;                         // this lane's key col
        const size_t kbase = ((size_t)b * LL + gN) * DD + h * DH;
        v2f b0, b1;
        b0.x = Ks[kbase + kb2];     b0.y = Ks[kbase + kb2 + 1];
        b1.x = Ks[kbase + 4 + kb2]; b1.y = Ks[kbase + 4 + kb2 + 1];

        v8f c = {0.f, 0.f, 0.f, 0.f, 0.f, 0.f, 0.f, 0.f};
        c = __builtin_amdgcn_wmma_f32_16x16x4_f32(false, a0, false, b0,
                                                  (short)0, c, false, false);
        c = __builtin_amdgcn_wmma_f32_16x16x4_f32(false, a1, false, b1,
                                                  (short)0, c, false, false);
        const int col = kt * 16 + Mn;
#pragma unroll
        for (int r = 0; r < 8; ++r)
            s_scores[r + 8 * hi][col] = c[r];
    }
    __syncthreads();

    // -------- Phase 2: fused  s += w + mask  (coalesced 128-bit NT loads)
    //                   + per-thread row-max partial
    const int rr = tid >> 4;
    const int cc = (tid & 15) * 4;
    {
        const v4f* w4 = (const v4f*)(wsrc + wbase);
        const v4f* m4 = (const v4f*)(msk + wbase);
        float mx = -INFINITY;
        for (int j = cc; j < LL; j += 64) {
            const int idx = (rr * LL + j) >> 2;
            const v4f wv = __builtin_nontemporal_load(&w4[idx]);
            const v4f mv = __builtin_nontemporal_load(&m4[idx]);
            v4f sv = *(const v4f*)&s_scores[rr][j];
            sv.x += wv.x + mv.x; sv.y += wv.y + mv.y;
            sv.z += wv.z + mv.z; sv.w += wv.w + mv.w;
            *(v4f*)&s_scores[rr][j] = sv;
            mx = fmaxf(mx, fmaxf(fmaxf(sv.x, sv.y), fmaxf(sv.z, sv.w)));
        }
        s_red[rr][tid & 15] = mx;
    }
    __syncthreads();
    if (tid < QT) {
        float mx = -INFINITY;
#pragma unroll
        for (int j = 0; j < 16; ++j) mx = fmaxf(mx, s_red[tid][j]);
        s_rowstat[tid] = mx;
    }
    __syncthreads();

    // -------- Phase 3: exp(s - max) in-place (vectorized) + row sum
    {
        const float mxv = s_rowstat[rr];
        float sum = 0.f;
        for (int j = cc; j < LL; j += 64) {
            v4f sv = *(const v4f*)&s_scores[rr][j];
            v4f e;
            e.x = __expf(sv.x - mxv); e.y = __expf(sv.y - mxv);
            e.z = __expf(sv.z - mxv); e.w = __expf(sv.w - mxv);
            *(v4f*)&s_scores[rr][j] = e;
            sum += (e.x + e.y) + (e.z + e.w);
        }
        s_red[rr][tid & 15] = sum;
    }
    __syncthreads();
    if (tid < QT) {
        float sum = 0.f;
#pragma unroll
        for (int j = 0; j < 16; ++j) sum += s_red[tid][j];
        s_rowstat[tid] = 1.f / sum;
    }
    __syncthreads();

    // -------- Phase 4: normalize in LDS (vectorized)
    float* aout = attn_out + wbase;
    for (int i = tid; i < QT * (LL / 4); i += NTHR) {
        const int r = i >> 9, j = (i & 511) * 4;
        const float inv = s_rowstat[r];
        v4f v = *(const v4f*)&s_scores[r][j];
        v.x *= inv; v.y *= inv; v.z *= inv; v.w *= inv;
        *(v4f*)&s_scores[r][j] = v;
#if !defined(__gfx1250__)
        // fallback path (host pass / non-gfx1250): direct NT store
        __builtin_nontemporal_store(v, (v4f*)(aout + (size_t)i * 4));
#endif
    }
    __syncthreads();

#if defined(__gfx1250__)
    // -------- Phase 5: stream attn to HBM directly from LDS (async, no VGPR
    // staging); overlaps with the ctx WMMA phase below which only READS LDS.
    for (int i = tid; i < QT * (LL / 4); i += NTHR) {
        const int r = i >> 9, j = (i & 511) * 4;
        const uint32_t lds_off = (uint32_t)(uintptr_t)&s_scores[r][j];
        const uint64_t gaddr   = (uint64_t)(uintptr_t)(aout + (size_t)i * 4);
        asm volatile("global_store_async_from_lds_b128 %0, %1, off"
                     :: "v"(gaddr), "v"(lds_off) : "memory");
    }
#endif

    // -------- Phase 6: ctx = attn @ V via chained WMMA (V padded 8 -> 16 cols)
    {
        v8f c = {0.f, 0.f, 0.f, 0.f, 0.f, 0.f, 0.f, 0.f};
        const int N = lane & 15;
        const int kwbase = wave * (LL / NWAVE);          // 256 keys per wave
        for (int kc = 0; kc < (LL / NWAVE) / 4; ++kc) {  // 64 chained WMMAs
            const int k0 = kwbase + kc * 4 + kb2;
            v2f a;
            a.x = s_scores[Mn][k0];
            a.y = s_scores[Mn][k0 + 1];
            v2f bb;
            if (N < 8) {
                const size_t vb = ((size_t)b * LL + k0) * DD + h * DH + N;
                bb.x = Vs[vb];
                bb.y = Vs[vb + DD];
            } else {
                bb.x = 0.f; bb.y = 0.f;
            }
            c = __builtin_amdgcn_wmma_f32_16x16x4_f32(false, a, false, bb,
                                                      (short)0, c, false, false);
        }
        if (N < 8) {
#pragma unroll
            for (int r = 0; r < 8; ++r) s_ctx[wave][r + 8 * hi][N] = c[r];
        }
    }
    __syncthreads();

    // Reduce per-wave ctx partials, write to workspace [B, L, 16]
    if (tid < QT * DH) {
        const int mrow = tid >> 3, d = tid & 7;
        float acc = 0.f;
#pragma unroll
        for (int wv = 0; wv < NWAVE; ++wv) acc += s_ctx[wv][mrow][d];
        ctxws[((size_t)b * LL + q0 + mrow) * DD + h * DH + d] = acc;
    }

#if defined(__gfx1250__)
    // Drain async stores (S_ENDPGM also implies wait-idle; explicit for clarity)
    asm volatile("s_wait_asynccnt 0x0" ::: "memory");
#endif
}

// ---------------------------------------------------------------------------
// Kernel 3: out = ctx @ Wo^T + bo + LN(query). One thread per row.
// (q0 residual is recomputed here to avoid an extra 512 KB workspace buffer.)
// ---------------------------------------------------------------------------
__global__ __launch_bounds__(NTHR)
void out_kernel(const float* __restrict__ ctxws, const float* __restrict__ query,
                const float* __restrict__ ln_qg, const float* __restrict__ ln_qb,
                const float* __restrict__ Wo, const float* __restrict__ bo,
                float* __restrict__ out)
{
    __shared__ float sW[256];
    __shared__ float sb[16], sg[16], sbeta[16];
    const int tid = threadIdx.x;
    sW[tid] = Wo[tid];
    if (tid < 16) { sb[tid] = bo[tid]; sg[tid] = ln_qg[tid]; sbeta[tid] = ln_qb[tid]; }
    __syncthreads();

    const int row = blockIdx.x * NTHR + tid;   // 0..8191
    const float* xr = query + (size_t)row * DD;
    float x[16];
    float m = 0.f;
#pragma unroll
    for (int i = 0; i < 16; ++i) { x[i] = xr[i]; m += x[i]; }
    m *= (1.f / 16.f);
    float v = 0.f;
#pragma unroll
    for (int i = 0; i < 16; ++i) { float d = x[i] - m; v += d * d; }
    v *= (1.f / 16.f);
    const float rinv = rsqrtf(v + 1e-5f);
#pragma unroll
    for (int i = 0; i < 16; ++i) x[i] = (x[i] - m) * rinv * sg[i] + sbeta[i];

    const float* cr = ctxws + (size_t)row * DD;
    float cx[16];
#pragma unroll
    for (int i = 0; i < 16; ++i) cx[i] = cr[i];

    float* orow = out + (size_t)row * DD;
#pragma unroll
    for (int j = 0; j < 16; ++j) {
        float acc = sb[j] + x[j];
#pragma unroll
        for (int i = 0; i < 16; ++i) acc += cx[i] * sW[j * 16 + i];
        orow[j] = acc;
    }
}

// ---------------------------------------------------------------------------
extern "C" void kernel_launch(void* const* d_in, const int* in_sizes, int n_in,
                              void* d_out, int out_size, void* d_ws, size_t ws_size,
                              hipStream_t stream) {
    const float* query = (const float*)d_in[0];
    const float* key   = (const float*)d_in[1];
    const float* value = (const float*)d_in[2];
    const float* w     = (const float*)d_in[3];
    const float* mask  = (const float*)d_in[4];
    const float* ln_qg = (const float*)d_in[5];
    const float* ln_qb = (const float*)d_in[6];
    const float* ln_kg = (const float*)d_in[7];
    const float* ln_kb = (const float*)d_in[8];
    const float* ln_vg = (const float*)d_in[9];
    const float* ln_vb = (const float*)d_in[10];
    const float* Wq    = (const float*)d_in[11];
    const float* bq    = (const float*)d_in[12];
    const float* Wk    = (const float*)d_in[13];
    const float* bk    = (const float*)d_in[14];
    const float* Wv    = (const float*)d_in[15];
    const float* bv    = (const float*)d_in[16];
    const float* Wo    = (const float*)d_in[17];
    const float* bo    = (const float*)d_in[18];

    const size_t rows = (size_t)BB * LL;        // 8192
    float* ws  = (float*)d_ws;
    float* Qs  = ws;
    float* Ks  = ws + rows * DD;                // +131072
    float* Vs  = ws + 2 * rows * DD;
    float* ctx = ws + 3 * rows * DD;            // total 2 MB scratch

    float* out  = (float*)d_out;
    float* attn = out + rows * DD;              // tuple: (out, attn) flat

    prep_kernel<<<96, NTHR, 0, stream>>>(query, key, value,
                                         ln_qg, ln_qb, ln_kg, ln_kb, ln_vg, ln_vb,
                                         Wq, bq, Wk, bk, Wv, bv,
                                         Qs, Ks, Vs);

    dim3 grid(LL / QT, HH, BB);                 // 128 x 2 x 4 workgroups
    attn_kernel<<<grid, NTHR, 0, stream>>>(Qs, Ks, Vs, w, mask, attn, ctx);

    out_kernel<<<32, NTHR, 0, stream>>>(ctx, query, ln_qg, ln_qb, Wo, bo, out);
}